// FWHTLayer_36429912604830
// MI455X (gfx1250) — compile-verified
//
#include <hip/hip_runtime.h>

// FWHT over N=4096 = 16^3 columns, 16384 rows, f32.
// y = x @ (H16 (x) H16 (x) H16) / 64, implemented as three mode-contractions
// by the 16x16 Hadamard matrix using V_WMMA_F32_16X16X4_F32 (K=4, chained x4).
// One wave32 owns one full row (16KB) in LDS -> no inter-wave barriers needed.
// Row I/O uses the CDNA5 async Global<->LDS path (ASYNCcnt-tracked).

typedef __attribute__((ext_vector_type(2))) float v2f;
typedef __attribute__((ext_vector_type(8))) float v8f;

#define N_COLS 4096
#define ROWS_PER_BLOCK 2   // 2 waves per block, 1 row per wave, 32KB static LDS

#if __has_builtin(__builtin_amdgcn_global_load_async_to_lds_b128)
#define HAVE_ASYNC_LOAD 1
#else
#define HAVE_ASYNC_LOAD 0
#endif
#if __has_builtin(__builtin_amdgcn_global_store_async_from_lds_b128)
#define HAVE_ASYNC_STORE 1
#else
#define HAVE_ASYNC_STORE 0
#endif

// Parameter type leaked by the clang diagnostic: pointer to
// 'int __attribute__((vector_size(4*sizeof(int))))' in AS(1)/AS(3).
typedef int v4i_vs __attribute__((vector_size(4 * sizeof(int))));
typedef __attribute__((address_space(1))) v4i_vs* gptr_b128;
typedef __attribute__((address_space(3))) v4i_vs* lptr_b128;

// Normalized Hadamard entry: H16[i][j] / 4 = +-0.25 (exact in f32).
// Three stages of 1/4 give the required 1/sqrt(4096) = 1/64 overall scale.
__device__ __forceinline__ float hq(int i, int j) {
  return (__popc(i & j) & 1) ? -0.25f : 0.25f;
}

__device__ __forceinline__ v8f wmma_acc(v2f a, v2f b, v8f c) {
  // 8 args: (neg_a, A, neg_b, B, c_mod, C, reuse_a, reuse_b)
  return __builtin_amdgcn_wmma_f32_16x16x4_f32(false, a, false, b, (short)0, c,
                                               false, false);
}

__global__ __launch_bounds__(32 * ROWS_PER_BLOCK)
void fwht4096_wmma(const float* __restrict__ x, float* __restrict__ y) {
  __shared__ float lds[ROWS_PER_BLOCK][N_COLS];

  const int wave = threadIdx.x >> 5;
  const int lane = threadIdx.x & 31;
  const int row  = blockIdx.x * ROWS_PER_BLOCK + wave;
  float* L = lds[wave];

  // Fragment coordinates (CDNA5 WMMA f32 VGPR layouts):
  //   A 16x4 : lane m = lane%16 holds K = 2*(lane/16) + {0,1} per VGPR pair
  //   B 4x16 : lane col n = lane%16, rows K = 2*(lane/16)+{0,1}
  //   C/D    : lane col n = lane%16, row = v + 8*(lane/16) for VGPR v
  const int half = lane >> 4;
  const int m    = lane & 15;
  const int n    = m;

  // ---------------- load one row (4096 f32) to LDS, coalesced 16B/lane ------
  {
    const float* src = x + (size_t)row * N_COLS;
#if HAVE_ASYNC_LOAD
    #pragma unroll
    for (int j = 0; j < 32; ++j) {
      const int o = j * 128 + lane * 4;
      __builtin_amdgcn_global_load_async_to_lds_b128(
          (gptr_b128)(src + o), (lptr_b128)(L + o), 0, 0);
    }
    // ASYNCcnt ops are unordered vs DS ops: drain before Stage A reads LDS.
    asm volatile("s_wait_asynccnt 0x0" ::: "memory");
#else
    #pragma unroll
    for (int j = 0; j < 32; ++j) {
      const int o = j * 128 + lane * 4;
      *(float4*)(L + o) = *(const float4*)(src + o);
    }
#endif
  }

  // Hadamard fragments when H is the A-matrix: haH[t] covers K = 4t+2*half+{0,1}
  // Hadamard fragments when H is the B-matrix: hbH[t] covers same K rows.
  v2f haH[4], hbH[4];
  #pragma unroll
  for (int t = 0; t < 4; ++t) {
    const int k = 4 * t + 2 * half;
    haH[t].x = hq(m, k);
    haH[t].y = hq(m, k + 1);
    hbH[t].x = hq(k, n);
    hbH[t].y = hq(k + 1, n);
  }

  // ---- Stage A: contract i0.  View row as M[256,16]; D = M_g @ H per 16-row
  // block g.  Data is the A-matrix, H is the B-matrix. ------------------------
  for (int g = 0; g < 16; ++g) {
    const int base = g * 256;  // rows 16g..16g+15, each 16 contiguous floats
    v8f c = {0.f, 0.f, 0.f, 0.f, 0.f, 0.f, 0.f, 0.f};
    #pragma unroll
    for (int t = 0; t < 4; ++t) {
      const int k = 4 * t + 2 * half;
      v2f a;
      a.x = L[base + m * 16 + k];
      a.y = L[base + m * 16 + k + 1];
      c = wmma_acc(a, hbH[t], c);
    }
    #pragma unroll
    for (int v = 0; v < 8; ++v)
      L[base + (v + 8 * half) * 16 + n] = c[v];
  }

  // ---- Stage B: contract i1.  For each i2, slice S[i1][i0]; D = H @ S.
  // H is the A-matrix, data slice is the B-matrix. ---------------------------
  for (int i2 = 0; i2 < 16; ++i2) {
    const int base = i2 * 256;
    v8f c = {0.f, 0.f, 0.f, 0.f, 0.f, 0.f, 0.f, 0.f};
    #pragma unroll
    for (int t = 0; t < 4; ++t) {
      const int k = 4 * t + 2 * half;
      v2f b;
      b.x = L[base + k * 16 + n];
      b.y = L[base + (k + 1) * 16 + n];
      c = wmma_acc(haH[t], b, c);
    }
    #pragma unroll
    for (int v = 0; v < 8; ++v)
      L[base + (v + 8 * half) * 16 + n] = c[v];
  }

  // ---- Stage C: contract i2.  For each i1, slice T[i2][i0]; D = H @ T. -----
  for (int i1 = 0; i1 < 16; ++i1) {
    v8f c = {0.f, 0.f, 0.f, 0.f, 0.f, 0.f, 0.f, 0.f};
    #pragma unroll
    for (int t = 0; t < 4; ++t) {
      const int k = 4 * t + 2 * half;
      v2f b;
      b.x = L[k * 256 + i1 * 16 + n];
      b.y = L[(k + 1) * 256 + i1 * 16 + n];
      c = wmma_acc(haH[t], b, c);
    }
    #pragma unroll
    for (int v = 0; v < 8; ++v)
      L[(v + 8 * half) * 256 + i1 * 16 + n] = c[v];
  }

  // ---------------- store row back, coalesced 16B/lane ----------------------
  {
    float* dst = y + (size_t)row * N_COLS;
#if HAVE_ASYNC_STORE
    // The async engine's LDS reads are unordered vs this wave's ds_stores:
    // drain DScnt first.  S_ENDPGM does an implicit wait-idle, so the async
    // stores themselves need no trailing wait.
    asm volatile("s_wait_dscnt 0x0" ::: "memory");
    #pragma unroll
    for (int j = 0; j < 32; ++j) {
      const int o = j * 128 + lane * 4;
      __builtin_amdgcn_global_store_async_from_lds_b128(
          (gptr_b128)(dst + o), (lptr_b128)(L + o), 0, 0);
    }
#else
    #pragma unroll
    for (int j = 0; j < 32; ++j) {
      const int o = j * 128 + lane * 4;
      *(float4*)(dst + o) = *(const float4*)(L + o);
    }
#endif
  }
}

extern "C" void kernel_launch(void* const* d_in, const int* in_sizes, int n_in,
                              void* d_out, int out_size, void* d_ws, size_t ws_size,
                              hipStream_t stream) {
  const float* x = (const float*)d_in[0];
  float* out = (float*)d_out;
  const int rows = in_sizes[0] / N_COLS;          // 16384
  const int grid = rows / ROWS_PER_BLOCK;         // 8192 blocks, 2 waves each
  fwht4096_wmma<<<grid, 32 * ROWS_PER_BLOCK, 0, stream>>>(x, out);
}